// MMCL_54159537603140
// MI455X (gfx1250) — compile-verified
//
#include <hip/hip_runtime.h>
#include <stdint.h>

#define NCOL 32768
#define PPOS 8
#define HNEG 327
#define TPB  512

// ---- LDS address helper: raw 32-bit LDS byte offset for async-to-LDS ops ----
typedef __attribute__((address_space(3))) uint8_t lds_u8;
__device__ __forceinline__ uint32_t lds_off(const void* p) {
  return (uint32_t)(uintptr_t)(lds_u8*)(p);
}

// Monotone float<->u32 key (descending float == descending key)
__device__ __forceinline__ uint32_t f2k(uint32_t u) {
  return (u & 0x80000000u) ? ~u : (u | 0x80000000u);
}
__device__ __forceinline__ float k2f(uint32_t k) {
  uint32_t u = (k & 0x80000000u) ? (k & 0x7fffffffu) : ~k;
  return __uint_as_float(u);
}

__global__ __launch_bounds__(TPB, 1)
void mmcl_row_kernel(const float* __restrict__ logits,
                     const uint8_t* __restrict__ targets,
                     float* __restrict__ row_loss) {
  extern __shared__ uint32_t keys[];   // NCOL x u32 (raw bits, then sort keys)
  __shared__ uint32_t hist[256];
  __shared__ float    fred[TPB];       // deterministic fixed-order reduction buffer
  __shared__ float    posv[PPOS];
  __shared__ uint32_t poscnt;
  __shared__ uint32_t negmaxkey;
  __shared__ uint32_t s_prefix;
  __shared__ uint32_t s_cntabove;

  const int tid = threadIdx.x;
  const int row = blockIdx.x;
  const float*   lg = logits  + (size_t)row * NCOL;
  const uint8_t* tg = targets + (size_t)row * NCOL;

  if (tid == 0) { poscnt = 0u; negmaxkey = 0u; }

  // ---- Phase 0: async bulk copy of the whole row into LDS (CDNA5 ASYNCcnt path).
  // 16 back-to-back b128 async loads per thread, one wait, then workgroup barrier.
  {
    const uint32_t base = lds_off(keys);
#pragma unroll
    for (int c = 0; c < (NCOL / 4) / TPB; ++c) {
      const int i = tid + c * TPB;
      uint32_t la = base + (uint32_t)i * 16u;
      const float* ga = lg + (size_t)i * 4;
      asm volatile("global_load_async_to_lds_b128 %0, %1, off"
                   :: "v"(la), "v"(ga) : "memory");
    }
    asm volatile("s_wait_asynccnt 0" ::: "memory");
  }
  __syncthreads();

  // ---- Phase 1: build keys, collect positives, max negative (integer atomics only) ----
  uint32_t local_negmax = 0u;
  for (int j = tid; j < NCOL; j += TPB) {
    uint32_t u = keys[j];
    uint32_t k = f2k(u);
    if (tg[j]) {
      uint32_t slot = atomicAdd(&poscnt, 1u);
      if (slot < PPOS) posv[slot] = __uint_as_float(u);
      keys[j] = 0u;                    // exclude positive from negative top-k
    } else {
      keys[j] = k;
      local_negmax = local_negmax > k ? local_negmax : k;
    }
  }
  atomicMax(&negmaxkey, local_negmax);
  __syncthreads();
  const float h0 = k2f(negmaxkey);     // largest negative logit (shift point)

  // ---- Phase 2: 4-level count-only radix select (u32 atomics -> deterministic) ----
  uint32_t prefix = 0u, cntabove = 0u;
  for (int level = 0; level < 4; ++level) {
    for (int b = tid; b < 256; b += TPB) hist[b] = 0u;
    __syncthreads();
    const int shL = 8 * (3 - level);
    for (int j = tid; j < NCOL; j += TPB) {
      uint32_t k  = keys[j];
      uint32_t hi = (uint32_t)(((uint64_t)k) >> (shL + 8));  // 0 at level 0
      if (hi == prefix) atomicAdd(&hist[(k >> shL) & 0xFFu], 1u);
    }
    __syncthreads();
    if (tid == 0) {
      uint32_t need = HNEG - cntabove;
      uint32_t cum = 0u;
      int b = 255;
      for (; b > 0; --b) {
        uint32_t c = hist[b];
        if (cum + c >= need) break;
        cum += c;
      }
      s_prefix   = (prefix << 8) | (uint32_t)b;
      s_cntabove = cntabove + cum;
    }
    __syncthreads();
    prefix   = s_prefix;
    cntabove = s_cntabove;
  }
  const uint32_t thr = prefix;          // exact 32-bit key of the H-th largest negative
                                        // cntabove = #elements with key strictly > thr

  // ---- Phase 3: deterministic exp-sum over keys > thr (fixed-order tree reduce) ----
  float part = 0.0f;
  for (int j = tid; j < NCOL; j += TPB) {
    uint32_t k = keys[j];
    if (k > thr) part += expf(10.0f * (k2f(k) - h0));
  }
  fred[tid] = part;
  __syncthreads();
  for (int off = TPB / 2; off > 0; off >>= 1) {
    if (tid < off) fred[tid] += fred[tid + off];
    __syncthreads();
  }

  // ---- Phase 4: thread 0 assembles the row loss ----
  if (tid == 0) {
    const float thr_val = k2f(thr);
    float sneg = fred[0]
               + (float)(HNEG - cntabove) * expf(10.0f * (thr_val - h0)); // exact ties

    float s[PPOS];
    for (int i = 0; i < PPOS; ++i) s[i] = posv[i];
    for (int i = 1; i < PPOS; ++i) {                        // 8-elem insertion sort, desc
      float v = s[i]; int j = i - 1;
      while (j >= 0 && s[j] < v) { s[j + 1] = s[j]; --j; }
      s[j + 1] = v;
    }

    float acc = 0.0f;
    for (int i = 0; i < PPOS; ++i) {
      int mj = i < (PPOS - i) ? i : (PPOS - i);             // largest positive in row i
      float mi = fmaxf(s[mj], h0);                          // per-row-i shift: all exps <= 0
      float a = (float)(PPOS - i) * expf(10.0f * (s[i] - mi));
      for (int j = PPOS - i; j < PPOS; ++j) a += expf(10.0f * (s[j] - mi));
      a += sneg * expf(10.0f * (h0 - mi));
      acc += 10.0f * mi + logf(a) - 10.0f * s[i];           // LSE_i - 10*s[i]
    }
    row_loss[row] = acc / (float)PPOS;
  }
}

__global__ __launch_bounds__(256)
void mmcl_reduce_kernel(const float* __restrict__ row_loss, float* __restrict__ out, int B) {
  __shared__ float red[256];
  const int t = threadIdx.x;
  float s = 0.0f;
  for (int i = t; i < B; i += 256) s += row_loss[i];        // fixed order: deterministic
  red[t] = s;
  __syncthreads();
  for (int off = 128; off > 0; off >>= 1) {
    if (t < off) red[t] += red[t + off];
    __syncthreads();
  }
  if (t == 0) out[0] = red[0] / (float)B;
}

extern "C" void kernel_launch(void* const* d_in, const int* in_sizes, int n_in,
                              void* d_out, int out_size, void* d_ws, size_t ws_size,
                              hipStream_t stream) {
  const float*   logits  = (const float*)d_in[0];
  const uint8_t* targets = (const uint8_t*)d_in[1];
  float* rl  = (float*)d_ws;
  float* out = (float*)d_out;
  const int B = in_sizes[0] / NCOL;
  const size_t shmem = (size_t)NCOL * sizeof(uint32_t);    // 128 KB row buffer in LDS
  hipFuncSetAttribute((const void*)mmcl_row_kernel,
                      hipFuncAttributeMaxDynamicSharedMemorySize, (int)shmem);
  mmcl_row_kernel<<<B, TPB, shmem, stream>>>(logits, targets, rl);
  mmcl_reduce_kernel<<<1, 256, 0, stream>>>(rl, out, B);
}